// RWKV_RNN_64287070486921
// MI455X (gfx1250) — compile-verified
//
#include <hip/hip_runtime.h>
#include <hip/hip_bf16.h>
#include <cstddef>

// RWKV single-token inference for MI455X (gfx1250, wave32).
// Memory-bound (~1.5 GB fp32 weights/token, 23.3 TB/s => ~65us floor).
// All matvecs run through V_WMMA_F32_16X16X4_F32: one wave owns 16 output
// rows, B operand = x broadcast into all 16 columns (compute is free under
// the HBM roofline). b128 global loads stream weights; x lives in LDS.

#define D_DIM 1024
#define H_DIM 4096
#define V_DIM 50257
#define L_DIM 24

typedef __attribute__((ext_vector_type(2))) float v2f;
typedef __attribute__((ext_vector_type(8))) float v8f;

// ---------------------------------------------------------------------------
// Core: 16-row fp32 matvec tile via V_WMMA_F32_16X16X4_F32.
// Lane l (wave32): m = l&15 selects the output row, half = l>>4 selects which
// 4-float K-chunk of the 8-float macro step this lane covers. A and B use the
// identical position->K mapping, so the MAC pairing W[m,k]*x[k] is correct
// independent of the hardware's internal K permutation.
// ---------------------------------------------------------------------------
__device__ __forceinline__ v8f wmma_dot16(const float* __restrict__ Wr,
                                          const float* __restrict__ xs,
                                          int K, int half) {
  v8f acc = {0.f, 0.f, 0.f, 0.f, 0.f, 0.f, 0.f, 0.f};
  const int off = 4 * half;
#pragma unroll 4
  for (int k0 = 0; k0 < K; k0 += 8) {
    const float4 w4 = *reinterpret_cast<const float4*>(Wr + k0 + off);
    const float4 x4 = *reinterpret_cast<const float4*>(xs + k0 + off);
    v2f a0 = {w4.x, w4.y};  v2f b0 = {x4.x, x4.y};
    v2f a1 = {w4.z, w4.w};  v2f b1 = {x4.z, x4.w};
    acc = __builtin_amdgcn_wmma_f32_16x16x4_f32(false, a0, false, b0,
                                                (short)0, acc, false, false);
    acc = __builtin_amdgcn_wmma_f32_16x16x4_f32(false, a1, false, b1,
                                                (short)0, acc, false, false);
  }
  return acc;
}

__device__ __forceinline__ float block_reduce_sum(float v, float* red) {
  const int tid = threadIdx.x;
  red[tid] = v;
  __syncthreads();
  for (int s = blockDim.x >> 1; s > 0; s >>= 1) {
    if (tid < s) red[tid] += red[tid + s];
    __syncthreads();
  }
  const float r = red[0];
  __syncthreads();
  return r;
}

// ---------------------------------------------------------------------------
// x = layer_norm(emb[token], ln0)
// ---------------------------------------------------------------------------
__global__ void __launch_bounds__(256)
k_embed_ln0(const int* __restrict__ token, const float* __restrict__ emb,
            const float* __restrict__ w, const float* __restrict__ b,
            float* __restrict__ xbuf) {
  __shared__ __align__(16) float xs[D_DIM];
  __shared__ float red[256];
  const int tid = threadIdx.x;
  const float* e = emb + (size_t)token[0] * D_DIM;
  float s = 0.f;
  for (int d = tid; d < D_DIM; d += 256) { float v = e[d]; xs[d] = v; s += v; }
  s = block_reduce_sum(s, red);
  const float mean = s * (1.f / D_DIM);
  float s2 = 0.f;
  for (int d = tid; d < D_DIM; d += 256) { float t = xs[d] - mean; s2 += t * t; }
  s2 = block_reduce_sum(s2, red);
  const float rstd = rsqrtf(s2 * (1.f / D_DIM) + 1e-5f);
  for (int d = tid; d < D_DIM; d += 256)
    xbuf[d] = (xs[d] - mean) * rstd * w[d] + b[d];
}

// ---------------------------------------------------------------------------
// Stage 1: xn = LN(x,ln1); xk/xv/xr = mix(xn, state_sp); k/v/rpre = W @ mix.
// Grid: 3 matrices x 32 blocks; each block redundantly recomputes LN+mix
// (8 KB of LDS work) then streams 32 rows of its 4 MB weight slice.
// ---------------------------------------------------------------------------
__global__ void __launch_bounds__(64)
k_att_kvr(const float* __restrict__ xbuf,
          const float* __restrict__ ln1w, const float* __restrict__ ln1b,
          const float* __restrict__ tmk, const float* __restrict__ tmv,
          const float* __restrict__ tmr, const float* __restrict__ sp,
          const float* __restrict__ kw, const float* __restrict__ vw,
          const float* __restrict__ rw,
          float* __restrict__ kout, float* __restrict__ vout,
          float* __restrict__ rout, float* __restrict__ xn_state_out) {
  __shared__ __align__(16) float xs[D_DIM];
  __shared__ __align__(16) float mix[D_DIM];
  __shared__ float red[64];
  const int tid = threadIdx.x;
  const int matId = blockIdx.x >> 5;  // 0=k 1=v 2=r
  const int blk = blockIdx.x & 31;

  float s = 0.f;
  for (int d = tid; d < D_DIM; d += 64) { float v = xbuf[d]; xs[d] = v; s += v; }
  s = block_reduce_sum(s, red);
  const float mean = s * (1.f / D_DIM);
  float s2 = 0.f;
  for (int d = tid; d < D_DIM; d += 64) { float t = xs[d] - mean; s2 += t * t; }
  s2 = block_reduce_sum(s2, red);
  const float rstd = rsqrtf(s2 * (1.f / D_DIM) + 1e-5f);

  const float* tm = (matId == 0) ? tmk : (matId == 1 ? tmv : tmr);
  const bool wst = (matId == 0) && (blk == 0);
  for (int d = tid; d < D_DIM; d += 64) {
    const float xn = (xs[d] - mean) * rstd * ln1w[d] + ln1b[d];
    if (wst) xn_state_out[d] = xn;  // new_state[l,1,:]
    const float t = tm[d];
    mix[d] = xn * t + sp[d] * (1.f - t);
  }
  __syncthreads();

  const float* W = (matId == 0) ? kw : (matId == 1 ? vw : rw);
  float* out = (matId == 0) ? kout : (matId == 1 ? vout : rout);
  const int lane = tid & 31, half = lane >> 4, m = lane & 15, wave = tid >> 5;
  const int row_base = blk * 32 + wave * 16;
  const v8f acc = wmma_dot16(W + (size_t)(row_base + m) * D_DIM, mix, D_DIM, half);
  if (m == 0) {
    const int ob = row_base + 8 * half;
#pragma unroll
    for (int j = 0; j < 8; ++j) out[ob + j] = acc[j];
  }
}

// ---------------------------------------------------------------------------
// Stage 2: WKV recurrence (redundant per block, elementwise over D) into LDS,
// then x += ow @ (r * a/b). Block 0 also emits new aa/bb/pp state.
// ---------------------------------------------------------------------------
__global__ void __launch_bounds__(64)
k_att_out(const float* __restrict__ kbuf, const float* __restrict__ vbuf,
          const float* __restrict__ rpre, const float* __restrict__ tf,
          const float* __restrict__ td, const float* __restrict__ aaS,
          const float* __restrict__ bbS, const float* __restrict__ ppS,
          const float* __restrict__ ow, float* __restrict__ xbuf,
          float* __restrict__ aaO, float* __restrict__ bbO,
          float* __restrict__ ppO) {
  __shared__ __align__(16) float rab[D_DIM];
  const int tid = threadIdx.x;
  const bool wst = (blockIdx.x == 0);
  for (int d = tid; d < D_DIM; d += 64) {
    const float k = kbuf[d], v = vbuf[d];
    const float r = 1.f / (1.f + expf(-rpre[d]));
    const float aa = aaS[d], bb = bbS[d], pp = ppS[d];
    const float ww = tf[d] + k;
    const float q = fmaxf(pp, ww);
    const float e1 = expf(pp - q), e2 = expf(ww - q);
    rab[d] = r * ((e1 * aa + e2 * v) / (e1 * bb + e2));
    const float ww2 = pp + td[d];
    const float q2 = fmaxf(ww2, k);
    const float e1n = expf(ww2 - q2), e2n = expf(k - q2);
    if (wst) {
      aaO[d] = e1n * aa + e2n * v;
      bbO[d] = e1n * bb + e2n;
      ppO[d] = q2;
    }
  }
  __syncthreads();
  const int lane = tid & 31, half = lane >> 4, m = lane & 15, wave = tid >> 5;
  const int row_base = blockIdx.x * 32 + wave * 16;
  const v8f acc = wmma_dot16(ow + (size_t)(row_base + m) * D_DIM, rab, D_DIM, half);
  if (m == 0) {
    const int ob = row_base + 8 * half;
#pragma unroll
    for (int j = 0; j < 8; ++j) xbuf[ob + j] = xbuf[ob + j] + acc[j];
  }
}

// ---------------------------------------------------------------------------
// Stage 3: xn2 = LN(x,ln2); fk = relu(fkw@fxk)^2 (H rows); fr = sigmoid(frw@fxr).
// Blocks [0,128) -> fkw rows, [128,160) -> frw rows.
// ---------------------------------------------------------------------------
__global__ void __launch_bounds__(64)
k_ffn_kr(const float* __restrict__ xbuf,
         const float* __restrict__ ln2w, const float* __restrict__ ln2b,
         const float* __restrict__ ftmk, const float* __restrict__ ftmr,
         const float* __restrict__ fp, const float* __restrict__ fkw,
         const float* __restrict__ frw, float* __restrict__ fkout,
         float* __restrict__ frout, float* __restrict__ xn2_state_out) {
  __shared__ __align__(16) float xs[D_DIM];
  __shared__ __align__(16) float mix[D_DIM];
  __shared__ float red[64];
  const int tid = threadIdx.x;
  const bool isK = blockIdx.x < (H_DIM / 32);
  const int blk = isK ? blockIdx.x : (blockIdx.x - H_DIM / 32);

  float s = 0.f;
  for (int d = tid; d < D_DIM; d += 64) { float v = xbuf[d]; xs[d] = v; s += v; }
  s = block_reduce_sum(s, red);
  const float mean = s * (1.f / D_DIM);
  float s2 = 0.f;
  for (int d = tid; d < D_DIM; d += 64) { float t = xs[d] - mean; s2 += t * t; }
  s2 = block_reduce_sum(s2, red);
  const float rstd = rsqrtf(s2 * (1.f / D_DIM) + 1e-5f);

  const float* tm = isK ? ftmk : ftmr;
  const bool wst = (blockIdx.x == 0);
  for (int d = tid; d < D_DIM; d += 64) {
    const float xn2 = (xs[d] - mean) * rstd * ln2w[d] + ln2b[d];
    if (wst) xn2_state_out[d] = xn2;  // new_state[l,0,:]
    const float t = tm[d];
    mix[d] = xn2 * t + fp[d] * (1.f - t);
  }
  __syncthreads();

  const int lane = tid & 31, half = lane >> 4, m = lane & 15, wave = tid >> 5;
  const int row_base = blk * 32 + wave * 16;
  const float* W = isK ? fkw : frw;
  const v8f acc = wmma_dot16(W + (size_t)(row_base + m) * D_DIM, mix, D_DIM, half);
  if (m == 0) {
    const int ob = row_base + 8 * half;
#pragma unroll
    for (int j = 0; j < 8; ++j) {
      const float y = acc[j];
      if (isK) {
        const float rl = fmaxf(y, 0.f);
        fkout[ob + j] = rl * rl;
      } else {
        frout[ob + j] = 1.f / (1.f + expf(-y));
      }
    }
  }
}

// ---------------------------------------------------------------------------
// Stage 4: x += fr * (fvw @ fk), K = 4096 (fk staged in 16 KB of LDS).
// ---------------------------------------------------------------------------
__global__ void __launch_bounds__(64)
k_ffn_out(const float* __restrict__ fkbuf, const float* __restrict__ frbuf,
          const float* __restrict__ fvw, float* __restrict__ xbuf) {
  __shared__ __align__(16) float fks[H_DIM];
  const int tid = threadIdx.x;
  for (int d = tid; d < H_DIM; d += 64) fks[d] = fkbuf[d];
  __syncthreads();
  const int lane = tid & 31, half = lane >> 4, m = lane & 15, wave = tid >> 5;
  const int row_base = blockIdx.x * 32 + wave * 16;
  const v8f acc = wmma_dot16(fvw + (size_t)(row_base + m) * H_DIM, fks, H_DIM, half);
  if (m == 0) {
    const int ob = row_base + 8 * half;
#pragma unroll
    for (int j = 0; j < 8; ++j) {
      const int r = ob + j;
      xbuf[r] = xbuf[r] + frbuf[r] * acc[j];
    }
  }
}

// ---------------------------------------------------------------------------
// Head: logits = head @ LN(x, ln_out). 1571 blocks x 32 rows (tail clamped).
// ---------------------------------------------------------------------------
__global__ void __launch_bounds__(64)
k_head(const float* __restrict__ xbuf, const float* __restrict__ lnw,
       const float* __restrict__ lnb, const float* __restrict__ head,
       float* __restrict__ logits) {
  __shared__ __align__(16) float xs[D_DIM];
  __shared__ __align__(16) float xn[D_DIM];
  __shared__ float red[64];
  const int tid = threadIdx.x;
  float s = 0.f;
  for (int d = tid; d < D_DIM; d += 64) { float v = xbuf[d]; xs[d] = v; s += v; }
  s = block_reduce_sum(s, red);
  const float mean = s * (1.f / D_DIM);
  float s2 = 0.f;
  for (int d = tid; d < D_DIM; d += 64) { float t = xs[d] - mean; s2 += t * t; }
  s2 = block_reduce_sum(s2, red);
  const float rstd = rsqrtf(s2 * (1.f / D_DIM) + 1e-5f);
  for (int d = tid; d < D_DIM; d += 64)
    xn[d] = (xs[d] - mean) * rstd * lnw[d] + lnb[d];
  __syncthreads();

  const int lane = tid & 31, half = lane >> 4, m = lane & 15, wave = tid >> 5;
  const int row_base = blockIdx.x * 32 + wave * 16;
  int row = row_base + m;
  if (row >= V_DIM) row = V_DIM - 1;  // clamp tail: redundant compute, guarded store
  const v8f acc = wmma_dot16(head + (size_t)row * D_DIM, xn, D_DIM, half);
  if (m == 0) {
    const int ob = row_base + 8 * half;
#pragma unroll
    for (int j = 0; j < 8; ++j) {
      const int r = ob + j;
      if (r < V_DIM) logits[r] = acc[j];
    }
  }
}

// ---------------------------------------------------------------------------
extern "C" void kernel_launch(void* const* d_in, const int* in_sizes, int n_in,
                              void* d_out, int out_size, void* d_ws,
                              size_t ws_size, hipStream_t stream) {
  (void)in_sizes; (void)n_in; (void)out_size; (void)ws_size;
  const int*   token = (const int*)d_in[0];
  const float* emb   = (const float*)d_in[1];
  const float* ln0w  = (const float*)d_in[2];
  const float* ln0b  = (const float*)d_in[3];
  const float* ln1w  = (const float*)d_in[4];
  const float* ln1b  = (const float*)d_in[5];
  const float* ln2w  = (const float*)d_in[6];
  const float* ln2b  = (const float*)d_in[7];
  const float* tmk   = (const float*)d_in[8];
  const float* tmv   = (const float*)d_in[9];
  const float* tmr   = (const float*)d_in[10];
  const float* tf    = (const float*)d_in[11];
  const float* td    = (const float*)d_in[12];
  const float* kw    = (const float*)d_in[13];
  const float* vw    = (const float*)d_in[14];
  const float* rw    = (const float*)d_in[15];
  const float* ow    = (const float*)d_in[16];
  const float* ftmk  = (const float*)d_in[17];
  const float* ftmr  = (const float*)d_in[18];
  const float* fkw   = (const float*)d_in[19];
  const float* fvw   = (const float*)d_in[20];
  const float* frw   = (const float*)d_in[21];
  const float* lnow  = (const float*)d_in[22];
  const float* lnob  = (const float*)d_in[23];
  const float* head  = (const float*)d_in[24];
  const float* state = (const float*)d_in[25];

  float* out     = (float*)d_out;          // logits [V] ++ new_state [L,5,D]
  float* st_out  = out + V_DIM;
  float* ws      = (float*)d_ws;
  float* xbuf    = ws;                     // D
  float* kbuf    = ws + 1 * D_DIM;         // D
  float* vbuf    = ws + 2 * D_DIM;         // D
  float* rpre    = ws + 3 * D_DIM;         // D
  float* fkbuf   = ws + 4 * D_DIM;         // H
  float* frbuf   = ws + 4 * D_DIM + H_DIM; // D

  const size_t DD = (size_t)D_DIM * D_DIM;
  const size_t HD = (size_t)H_DIM * D_DIM;

  k_embed_ln0<<<1, 256, 0, stream>>>(token, emb, ln0w, ln0b, xbuf);

  for (int l = 0; l < L_DIM; ++l) {
    const float* stL  = state  + (size_t)l * 5 * D_DIM;
    float*       stoL = st_out + (size_t)l * 5 * D_DIM;
    k_att_kvr<<<96, 64, 0, stream>>>(
        xbuf, ln1w + l * D_DIM, ln1b + l * D_DIM, tmk + l * D_DIM,
        tmv + l * D_DIM, tmr + l * D_DIM, stL + 1 * D_DIM, kw + l * DD,
        vw + l * DD, rw + l * DD, kbuf, vbuf, rpre, stoL + 1 * D_DIM);
    k_att_out<<<32, 64, 0, stream>>>(
        kbuf, vbuf, rpre, tf + l * D_DIM, td + l * D_DIM, stL + 2 * D_DIM,
        stL + 3 * D_DIM, stL + 4 * D_DIM, ow + l * DD, xbuf, stoL + 2 * D_DIM,
        stoL + 3 * D_DIM, stoL + 4 * D_DIM);
    k_ffn_kr<<<160, 64, 0, stream>>>(
        xbuf, ln2w + l * D_DIM, ln2b + l * D_DIM, ftmk + l * D_DIM,
        ftmr + l * D_DIM, stL + 0 * D_DIM, fkw + (size_t)l * HD,
        frw + l * DD, fkbuf, frbuf, stoL + 0 * D_DIM);
    k_ffn_out<<<32, 64, 0, stream>>>(fkbuf, frbuf, fvw + (size_t)l * HD, xbuf);
  }

  k_head<<<(V_DIM + 31) / 32, 64, 0, stream>>>(xbuf, lnow, lnob, head, out);
}